// PGN_39453569581088
// MI455X (gfx1250) — compile-verified
//
#include <hip/hip_runtime.h>

// Pointer-generator final distribution for MI455X (gfx1250, wave32).
// Bandwidth-bound: ~385 MB moved, ~16.5 us floor at 23.3 TB/s.
// Strategy: one 256-thread block per (t,b) row; full extended row accumulated
// in LDS (320KB/WGP easily holds 30080 floats), scatter via ds_add_f32,
// streaming fused multiply-add with nontemporal B64 global traffic.

typedef float v2f __attribute__((ext_vector_type(2)));
typedef float v4f __attribute__((ext_vector_type(4)));

// Address-space-qualified pointee types for the gfx1250 async LDS builtins:
// clang prints AS1 (global) as '__device__ int *' in HIP mode.
typedef __attribute__((address_space(1))) int as1int;
typedef __attribute__((address_space(3))) int as3int;

constexpr int T_DEC   = 50;
constexpr int BATCH   = 32;
constexpr int VOCAB   = 30000;
constexpr int ENC_LEN = 200;
constexpr int MAX_OOV = 50;
constexpr int EXT     = VOCAB + MAX_OOV;   // 30050
constexpr int EXT_PAD = 30080;             // round up to multiple of 64 floats
constexpr int NTHREADS = 256;              // 8 wave32 waves

// LDS layout (floats): [0,EXT_PAD) accumulator ; then ENC_LEN attn ; ENC_LEN idx
constexpr size_t SMEM_BYTES = (size_t)(EXT_PAD + ENC_LEN + ENC_LEN) * sizeof(float);

__global__ __launch_bounds__(NTHREADS)
void pgn_final_dist(const float* __restrict__ pred,
                    const float* __restrict__ attn,
                    const float* __restrict__ pgens,
                    const int*   __restrict__ extinp,
                    float*       __restrict__ out) {
    extern __shared__ float smem[];
    float* acc    = smem;
    float* attn_s = smem + EXT_PAD;
    int*   idx_s  = (int*)(smem + EXT_PAD + ENC_LEN);

    const int tid = threadIdx.x;
    const int t   = blockIdx.x / BATCH;
    const int b   = blockIdx.x - t * BATCH;

    const float* pred_row = pred   + ((size_t)t * BATCH + b) * VOCAB;
    const float* attn_row = attn   + ((size_t)t * BATCH + b) * ENC_LEN; // attentions[:-1]
    const int*   idx_row  = extinp + (size_t)b * ENC_LEN;
    float*       out_row  = out    + ((size_t)b * T_DEC + t) * EXT;

    // ---- Stage attention + scatter indices into LDS (async on gfx1250) ----
#if __has_builtin(__builtin_amdgcn_global_load_async_to_lds_b32)
    if (tid < ENC_LEN) {
        __builtin_amdgcn_global_load_async_to_lds_b32(
            (as1int*)(attn_row + tid), (as3int*)(attn_s + tid), 0, 0);
        __builtin_amdgcn_global_load_async_to_lds_b32(
            (as1int*)(idx_row + tid), (as3int*)(idx_s + tid), 0, 0);
    }
#else
    if (tid < ENC_LEN) {
        attn_s[tid] = attn_row[tid];
        idx_s[tid]  = idx_row[tid];
    }
#endif

    // ---- Zero the extended-row accumulator with 128-bit LDS stores ----
    v4f* acc4 = (v4f*)acc;
    const v4f z = {0.0f, 0.0f, 0.0f, 0.0f};
    for (int i = tid; i < EXT_PAD / 4; i += NTHREADS) acc4[i] = z;

    const float pg = pgens[(size_t)t * BATCH + b];   // uniform -> scalar load

#if __has_builtin(__builtin_amdgcn_global_load_async_to_lds_b32)
#if __has_builtin(__builtin_amdgcn_s_wait_asynccnt)
    __builtin_amdgcn_s_wait_asynccnt(0);
#else
    asm volatile("s_wait_asynccnt 0" ::: "memory");
#endif
#endif
    __syncthreads();

    // ---- Scatter (1-pg)*attn into acc at extended indices: ds_add_f32 ----
    if (tid < ENC_LEN) {
        atomicAdd(&acc[idx_s[tid]], (1.0f - pg) * attn_s[tid]);
    }
    __syncthreads();

    // ---- Stream out = pg*pred + acc over [0,VOCAB), nontemporal B64 ----
    // (out rows are 8B-aligned: 30050*4 % 16 != 0, so float2 not float4)
    const v2f* pred2 = (const v2f*)pred_row;
    const v2f* acc2  = (const v2f*)acc;
    v2f*       out2  = (v2f*)out_row;
    for (int i = tid; i < VOCAB / 2; i += NTHREADS) {
        v2f p = __builtin_nontemporal_load(&pred2[i]);
        v2f a = acc2[i];
        v2f r;
        r.x = fmaf(pg, p.x, a.x);
        r.y = fmaf(pg, p.y, a.y);
        __builtin_nontemporal_store(r, &out2[i]);
    }

    // ---- OOV tail [VOCAB, EXT): scatter accumulator only ----
    if (tid < MAX_OOV) {
        __builtin_nontemporal_store(acc[VOCAB + tid], &out_row[VOCAB + tid]);
    }
}

extern "C" void kernel_launch(void* const* d_in, const int* in_sizes, int n_in,
                              void* d_out, int out_size, void* d_ws, size_t ws_size,
                              hipStream_t stream) {
    (void)in_sizes; (void)n_in; (void)out_size; (void)d_ws; (void)ws_size;

    const float* pred   = (const float*)d_in[0];   // (T,B,V) f32
    const float* attn   = (const float*)d_in[1];   // (T+1,B,L) f32
    const float* pgens  = (const float*)d_in[2];   // (T,B,1) f32
    const int*   extinp = (const int*)  d_in[3];   // (B,L) i32
    float*       out    = (float*)d_out;           // (B,T,EXT) f32

    // Allow >64KB dynamic LDS (WGP supports 320KB). Idempotent, capture-safe.
    (void)hipFuncSetAttribute((const void*)pgn_final_dist,
                              hipFuncAttributeMaxDynamicSharedMemorySize,
                              (int)SMEM_BYTES);

    pgn_final_dist<<<dim3(T_DEC * BATCH), dim3(NTHREADS), SMEM_BYTES, stream>>>(
        pred, attn, pgens, extinp, out);
}